// InteractionLigandEncoder_10058813407622
// MI455X (gfx1250) — compile-verified
//
#include <hip/hip_runtime.h>
#include <hip/hip_bf16.h>

typedef float v2f __attribute__((ext_vector_type(2)));
typedef float v8f __attribute__((ext_vector_type(8)));

#define NB    256   // batches (one workgroup each)
#define NI    1024  // interactions per batch
#define NP    128   // param encoding dim (K of the gate matvec)
#define NH    128   // hidden dim
#define NROWS 128   // ATOMS_PER_MOL = output rows per batch
#define LPROT 1024

// One workgroup (256 threads = 8 wave32) per batch b.
// Phase 1: gates via V_WMMA_F32_16X16X4_F32 (16 rows/wave-tile, K-loop of 32).
// Phase 2: LDS scatter-accumulate with ds_add_f32.
// Phase 3: coalesced b128 stores of the 64 KB accumulator tile.
__launch_bounds__(256, 4)
__global__ void interaction_ligand_encoder_kernel(
    const float* __restrict__ param_enc,      // (B, I, P)
    const float* __restrict__ h_prot,         // (B, LPROT, H)
    const float* __restrict__ w1,             // (P, 1)
    const float* __restrict__ b1,             // (1,)
    const int*   __restrict__ indices_lig,    // (B, I)  in [0, NROWS)
    const int*   __restrict__ indices_prot,   // (B, I)  in [0, LPROT)
    const int*   __restrict__ protein_idx,    // (B,)
    const int*   __restrict__ lig_offsets,    // (B,)
    const unsigned char* __restrict__ mask,   // (B, I) bool
    float*       __restrict__ out)            // (B*NROWS, H)
{
    __shared__ float acc[NROWS * NH];   // 64 KB accumulator (one batch's output tile)
    __shared__ float gate[NI];          // 4 KB gates

    const int b    = blockIdx.x;
    const int tid  = threadIdx.x;       // 0..255
    const int wave = tid >> 5;          // 0..7
    const int lane = tid & 31;          // wave32

    // Zero the accumulator (output rows not hit by any index must be 0).
    for (int f = tid; f < NROWS * NH; f += 256) acc[f] = 0.0f;

    // ---------------- Phase 1: gate[i] = relu(param_enc[b,i,:] . w1 + b1) ----------------
    // WMMA f32 16x16x4: A = 16x4 tile of param_enc rows, B = 4x16 with every
    // column equal to the w1 chunk -> every column of D equals the dot product.
    const float bias = b1[0];
    const int   half = lane >> 4;       // 0: K pair {0,1}; 1: K pair {2,3}
    const int   m16  = lane & 15;       // A-row within the 16-row tile

    const float* pe_b = param_enc + (size_t)b * NI * NP;

    for (int tile = wave; tile < NI / 16; tile += 8) {
        const int    row0 = tile * 16;
        const float* rowp = pe_b + (size_t)(row0 + m16) * NP;

        // Prefetch this lane's row for the next tile handled by this wave.
        __builtin_prefetch(rowp + 8 * 16 * NP, 0, 1);

        v8f c = {};
        for (int k = 0; k < NP; k += 4) {
            const int kk = k + half * 2;
            v2f a, bb;
            a.x  = rowp[kk];
            a.y  = rowp[kk + 1];
            bb.x = w1[kk];
            bb.y = w1[kk + 1];
            // 8 args: (neg_a, A, neg_b, B, c_mod, C, reuse_a, reuse_b)
            c = __builtin_amdgcn_wmma_f32_16x16x4_f32(false, a, false, bb,
                                                      (short)0, c, false, false);
        }
        // D layout: lane L, element v -> row (v + 8*half); all 16 columns equal.
        if (m16 == 0) {
            #pragma unroll
            for (int v = 0; v < 8; ++v) {
                const int i = row0 + half * 8 + v;
                float g = c[v] + bias;
                g = g > 0.0f ? g : 0.0f;                       // relu
                if (!mask[(size_t)b * NI + i]) g = 0.0f;       // fold mask into gate
                gate[i] = g;
            }
        }
    }
    __syncthreads();

    // ---------------- Phase 2: scatter-accumulate into LDS ----------------
    // One i per wave per iteration: the 512 B h_prot row is loaded with 4
    // coalesced b32 loads/lane and accumulated with bank-conflict-free
    // ds_add_f32 (h = lane + 32*j -> consecutive lanes hit consecutive banks).
    const int    pidx = protein_idx[b];
    const float* hp_b = h_prot + (size_t)pidx * LPROT * NH;
    const int*   il   = indices_lig  + (size_t)b * NI;
    const int*   ip   = indices_prot + (size_t)b * NI;

    for (int i = wave; i < NI; i += 8) {
        const float g = gate[i];                // uniform across the wave
        if (g != 0.0f) {                        // relu zeros contribute nothing
            const int    r   = il[i];
            const float* row = hp_b + (size_t)ip[i] * NH;
            float*       dst = acc + r * NH;
            #pragma unroll
            for (int j = 0; j < 4; ++j) {
                const int h = lane + 32 * j;
                atomicAdd(&dst[h], g * row[h]); // ds_add_f32
            }
        }
    }
    __syncthreads();

    // ---------------- Phase 3: stream the 64 KB tile to global ----------------
    const int off = lig_offsets[b];             // == b * NROWS
    float* ob = out + (size_t)off * NH;
    for (int f4 = tid; f4 < (NROWS * NH) / 4; f4 += 256) {
        *(float4*)(ob + f4 * 4) = *(const float4*)(acc + f4 * 4);
    }
}

extern "C" void kernel_launch(void* const* d_in, const int* in_sizes, int n_in,
                              void* d_out, int out_size, void* d_ws, size_t ws_size,
                              hipStream_t stream) {
    const float*         param_enc    = (const float*)d_in[0];
    const float*         h_prot       = (const float*)d_in[1];
    const float*         w1           = (const float*)d_in[2];
    const float*         b1           = (const float*)d_in[3];
    const int*           indices_lig  = (const int*)d_in[4];
    const int*           indices_prot = (const int*)d_in[5];
    const int*           protein_idx  = (const int*)d_in[6];
    const int*           lig_offsets  = (const int*)d_in[7];
    const unsigned char* mask         = (const unsigned char*)d_in[8];
    float*               out          = (float*)d_out;

    interaction_ligand_encoder_kernel<<<NB, 256, 0, stream>>>(
        param_enc, h_prot, w1, b1, indices_lig, indices_prot,
        protein_idx, lig_offsets, mask, out);
}